// GraphAttentionLayer_88742614270406
// MI455X (gfx1250) — compile-verified
//
#include <hip/hip_runtime.h>
#include <hip/hip_bf16.h>

typedef float v2f __attribute__((ext_vector_type(2)));
typedef float v8f __attribute__((ext_vector_type(8)));

#define DIN   128
#define DOUT  64
#define KNBR  32
#define NEGV  (-9.0e15f)

// ---------------------------------------------------------------------------
// Kernel 1: h = x @ W^T + b  via V_WMMA_F32_16X16X4_F32 (exact f32 MACs).
// One wave per 16-row tile; 4 accumulators (16x16 each) cover all 64 columns.
// W staged to LDS as K-pair-packed transposed tiles: Wt2[k/2][n] = {W[n][k], W[n][k+1]}
// so each B fragment is one 8-byte LDS load.
// ---------------------------------------------------------------------------
__global__ __launch_bounds__(256) void gat_gemm_h(const float* __restrict__ x,
                                                  const float* __restrict__ W,
                                                  const float* __restrict__ bias,
                                                  float* __restrict__ h,
                                                  int n_rows, int n_tiles)
{
    __shared__ __align__(16) float Wt2[DIN / 2][DOUT][2]; // 32 KB
    __shared__ float bs[DOUT];

    const int t = threadIdx.x;

    // Stage W transposed (bank-conflict-free LDS writes: lanes sweep n).
    for (int i = t; i < DOUT * DIN; i += 256) {
        int k = i >> 6;          // 0..127
        int n = i & (DOUT - 1);  // 0..63
        Wt2[k >> 1][n][k & 1] = W[n * DIN + k];
    }
    if (t < DOUT) bs[t] = bias[t];
    __syncthreads();

    const int wave = t >> 5;
    const int lane = t & 31;
    const int half = lane >> 4;   // 0/1: selects K-pair within K=4 step
    const int l16  = lane & 15;

    const int tile = blockIdx.x * 8 + wave;   // wave-uniform
    if (tile >= n_tiles) return;              // whole-wave exit: EXEC stays all-1s

    // A fragment source row (clamped for a possible ragged last tile).
    int rowA = tile * 16 + l16;
    if (rowA >= n_rows) rowA = n_rows - 1;
    const float* __restrict__ xrow = x + (size_t)rowA * DIN + 2 * half;

    v8f c0 = {}, c1 = {}, c2 = {}, c3 = {};

    #pragma unroll 4
    for (int k = 0; k < DIN; k += 4) {
        v2f a;
        a.x = xrow[k];
        a.y = xrow[k + 1];
        const int kp = (k >> 1) + half;       // packed K-pair index for this lane
        v2f b0 = *(const v2f*)&Wt2[kp][l16][0];
        v2f b1 = *(const v2f*)&Wt2[kp][16 + l16][0];
        v2f b2 = *(const v2f*)&Wt2[kp][32 + l16][0];
        v2f b3 = *(const v2f*)&Wt2[kp][48 + l16][0];
        // (neg_a, A, neg_b, B, c_mod, C, reuse_a, reuse_b)
        c0 = __builtin_amdgcn_wmma_f32_16x16x4_f32(false, a, false, b0, (short)0, c0, false, false);
        c1 = __builtin_amdgcn_wmma_f32_16x16x4_f32(false, a, false, b1, (short)0, c1, false, false);
        c2 = __builtin_amdgcn_wmma_f32_16x16x4_f32(false, a, false, b2, (short)0, c2, false, false);
        c3 = __builtin_amdgcn_wmma_f32_16x16x4_f32(false, a, false, b3, (short)0, c3, false, false);
    }

    // C/D layout: lane l, vgpr j -> M = j + 8*(l/16), N = l%16 (per 16-col block).
    const float bv0 = bs[l16], bv1 = bs[16 + l16], bv2 = bs[32 + l16], bv3 = bs[48 + l16];
    #pragma unroll
    for (int j = 0; j < 8; ++j) {
        const int m   = j + 8 * half;
        const int row = tile * 16 + m;
        if (row >= n_rows) continue;
        float o0 = c0[j] + bv0;
        float o1 = c1[j] + bv1;
        float o2 = c2[j] + bv2;
        float o3 = c3[j] + bv3;
        if (row == 0) { o0 = NEGV; o1 = NEGV; o2 = NEGV; o3 = NEGV; }  // sentinel row
        float* __restrict__ hr = h + (size_t)row * DOUT;
        hr[l16]      = o0;
        hr[16 + l16] = o1;
        hr[32 + l16] = o2;
        hr[48 + l16] = o3;
    }
}

// ---------------------------------------------------------------------------
// Kernel 2: per-node attention. One wave32 per node:
//   lane d pair-loads h[n], wave-reduce self_dot; lane k = neighbor k,
//   64-length dot via 16x b128 loads against LDS-broadcast h[n];
//   33-way softmax via shuffles; out = h * att[:,0]; out[0] = 0.
// ---------------------------------------------------------------------------
__global__ __launch_bounds__(256) void gat_attn(const float* __restrict__ h,
                                                const long long* __restrict__ a2a,
                                                float* __restrict__ out,
                                                int n_rows)
{
    __shared__ __align__(16) float hbuf[8][DOUT];   // per-wave slot, no cross-wave sync

    const int wave = threadIdx.x >> 5;
    const int lane = threadIdx.x & 31;
    const int n = blockIdx.x * 8 + wave;            // wave-uniform
    if (n >= n_rows) return;

    const float* __restrict__ hn = h + (size_t)n * DOUT;
    const float h0 = hn[2 * lane];
    const float h1 = hn[2 * lane + 1];
    hbuf[wave][2 * lane]     = h0;
    hbuf[wave][2 * lane + 1] = h1;

    // self_dot = |h[n]|^2 (wave reduce)
    float sd = h0 * h0 + h1 * h1;
    #pragma unroll
    for (int s = 16; s >= 1; s >>= 1) sd += __shfl_xor(sd, s, 32);

    // neighbor dot for this lane's neighbor
    const long long idx = a2a[(size_t)n * KNBR + lane];
    const float4* __restrict__ nb4 = (const float4*)(h + (size_t)idx * DOUT);
    const float4* __restrict__ hb4 = (const float4*)&hbuf[wave][0];
    float nd = 0.0f;
    #pragma unroll
    for (int d = 0; d < DOUT / 4; ++d) {
        const float4 a = nb4[d];
        const float4 c = hb4[d];          // same-address LDS broadcast across lanes
        nd += a.x * c.x + a.y * c.y + a.z * c.z + a.w * c.w;
    }

    // softmax over {sd, nd[0..31]} (max-subtracted, matching jax.nn.softmax)
    float mx = nd;
    #pragma unroll
    for (int s = 16; s >= 1; s >>= 1) mx = fmaxf(mx, __shfl_xor(mx, s, 32));
    mx = fmaxf(mx, sd);

    float e = expf(nd - mx);
    float es = e;
    #pragma unroll
    for (int s = 16; s >= 1; s >>= 1) es += __shfl_xor(es, s, 32);
    const float e0   = expf(sd - mx);
    float att0 = e0 / (e0 + es);
    if (n == 0) att0 = 0.0f;              // out.at[0].set(0)

    float* __restrict__ on = out + (size_t)n * DOUT;
    on[2 * lane]     = h0 * att0;
    on[2 * lane + 1] = h1 * att0;
}

extern "C" void kernel_launch(void* const* d_in, const int* in_sizes, int n_in,
                              void* d_out, int out_size, void* d_ws, size_t ws_size,
                              hipStream_t stream) {
    const float*     x    = (const float*)d_in[0];     // [N, 128] f32
    const long long* a2a  = (const long long*)d_in[1]; // [N, 32]  i64
    const float*     W    = (const float*)d_in[2];     // [64, 128] f32
    const float*     bias = (const float*)d_in[3];     // [64] f32
    float*           out  = (float*)d_out;             // [N, 64] f32

    const int n_rows  = in_sizes[0] / DIN;             // 100000
    const int n_tiles = (n_rows + 15) / 16;            // 6250
    float* h = (float*)d_ws;                           // [N, 64] scratch (25.6 MB)

    dim3 blk(256);
    gat_gemm_h<<<dim3((n_tiles + 7) / 8), blk, 0, stream>>>(x, W, bias, h, n_rows, n_tiles);
    gat_attn  <<<dim3((n_rows  + 7) / 8), blk, 0, stream>>>(h, a2a, out, n_rows);
}